// TemporalGraph_45818711113852
// MI455X (gfx1250) — compile-verified
//
#include <hip/hip_runtime.h>

// ---------------------------------------------------------------------------
// TemporalGraph pipeline for MI455X (gfx1250, wave32, WMMA):
//   conv1(Wd,k=3)+BN -> [distance/top-k block is provably dead code:
//   sim = -sqrt(d2)/norm <= 0 < 0.05 everywhere => idx == {0,1,2,3} always]
//   -> fixed-graph GCN (WMMA GEMM + analytic-degree fixup) -> conv2(Wu,k=3)+BN.
// All heavy math runs through V_WMMA_F32_16X16X4_F32 (f32-exact matrix op).
// B operands are staged in LDS pre-packed in WMMA fragment order so each
// fragment is a single ds_load_b64 into the operand register pair.
// ---------------------------------------------------------------------------

typedef __attribute__((ext_vector_type(2))) float v2f;
typedef __attribute__((ext_vector_type(4))) float v4f;
typedef __attribute__((ext_vector_type(8))) float v8f;

#define CC 256
#define HW 256
#define VV 128
#define BB 4
#define BCVH (CC * VV * HW)            // 8388608 elements per batch image

// Fragment-packed B tile: Xf[kk4(4)][j(16)][lane(32)] of float2 = 4096 floats
#define FRAG_FLOATS (4 * 16 * 32 * 2)

// One kernel for all three GEMM stages (strides parameterize layout/transpose):
//   OUT[b*BCVH + t*outT + co*outCo + p] =
//     scale[co]*( sum_k sum_ci W[co*wCo + ci*wCi + k*wTap]
//                   * IN[b*BCVH + (t+k+tapShift)*inT + ci*inCi + p] ) + shift[co]
// grid = (B*V, 2); block = 256 = 8 waves in a 2x4 grid; wave tile = 64co x 64p.
__global__ __launch_bounds__(256) void tg_gemm_tap(
    const float* __restrict__ IN, const float* __restrict__ W,
    const float* __restrict__ scale, const float* __restrict__ shift,
    float* __restrict__ OUT,
    int inTStride, int inCiStride,
    int outTStride, int outCoStride,
    int wCoStride, int wCiStride, int wTapStride,
    int ntaps, int tapShift)
{
    __shared__ __align__(16) float Xf[2 * FRAG_FLOATS];   // 32 KB, double buffered

    const int bt   = blockIdx.x;
    const int b    = bt / VV;
    const int t    = bt - b * VV;
    const int row0 = blockIdx.y * 128;
    const int tid  = threadIdx.x;
    const int wave = tid >> 5;
    const int lane = tid & 31;
    const int l16  = lane & 15;
    const int lhi  = lane >> 4;        // 0: K pair {0,1}, 1: K pair {2,3}
    const int wr   = wave >> 2;        // 0..1 : 64-row co block inside 128
    const int wc   = wave & 3;         // 0..3 : 64-col p block

    // ---- staging role of this thread: (kk4, klhi, j, h) -> row pair + 8 cols
    const int s_kk4 = tid >> 6;
    const int s_lhi = (tid >> 5) & 1;
    const int s_j   = (tid >> 1) & 15;
    const int s_h   = tid & 1;
    const int s_r0  = s_kk4 * 4 + s_lhi * 2;     // ci row pair within 16-chunk
    const int s_c   = s_j * 16 + s_h * 8;        // 8 source columns
    float* const s_dst =
        &Xf[((s_kk4 * 16 + s_j) * 32 + s_lhi * 16 + s_h * 8) * 2];

    v8f acc[16] = {};                  // acc[rt*4 + jc]

    const float* const INb  = IN + (long)b * BCVH;
    float* const OUTb       = OUT + (long)b * BCVH + (long)t * outTStride;
    // per-lane weight base: co = row0 + wr*64 + l16, K offset lhi*2
    const float* const Wl   = W + (long)(row0 + wr * 64 + l16) * wCoStride
                                + (long)(lhi * 2) * wCiStride;
    const float* const Bp   = &Xf[lane * 2];

    // ---- clip taps with out-of-range frames (wave-uniform, taps contiguous)
    int kstart = 0, kend = ntaps;
    while (kstart < kend && (t + kstart + tapShift) < 0) ++kstart;
    while (kend > kstart && (t + kend - 1 + tapShift) >= VV) --kend;
    const int nchunk = (kend - kstart) * 16;     // 16-ci chunks across taps

    v4f st[4];                                   // staged rows (r0,r0+1) x 8 cols

    auto load_chunk = [&](int m) {
        const int k  = kstart + (m >> 4);
        const int c0 = (m & 15) << 4;
        const int tt = t + k + tapShift;
        const float* p0 = INb + (long)tt * inTStride
                              + (long)(s_r0 + c0) * inCiStride + s_c;
        const float* p1 = p0 + inCiStride;
        st[0] = *(const v4f*)(p0);
        st[1] = *(const v4f*)(p0 + 4);
        st[2] = *(const v4f*)(p1);
        st[3] = *(const v4f*)(p1 + 4);
    };
    auto store_chunk = [&](int buf) {            // interleave rows -> float2 frags
        float* d = s_dst + buf * FRAG_FLOATS;
        v4f o;
        o.x = st[0].x; o.y = st[2].x; o.z = st[0].y; o.w = st[2].y;
        *(v4f*)(d + 0)  = o;
        o.x = st[0].z; o.y = st[2].z; o.z = st[0].w; o.w = st[2].w;
        *(v4f*)(d + 4)  = o;
        o.x = st[1].x; o.y = st[3].x; o.z = st[1].y; o.w = st[3].y;
        *(v4f*)(d + 8)  = o;
        o.x = st[1].z; o.y = st[3].z; o.z = st[1].w; o.w = st[3].w;
        *(v4f*)(d + 12) = o;
    };
    auto compute_chunk = [&](int m, int buf) {
        const int k  = kstart + (m >> 4);
        const int c0 = (m & 15) << 4;
        const float* Wck = Wl + (long)c0 * wCiStride + (long)k * wTapStride;
        const float* Bb  = Bp + buf * FRAG_FLOATS + (wc * 4) * 64;
        #pragma unroll
        for (int kk4 = 0; kk4 < 4; ++kk4) {
            v2f a[4], bv[4];
            #pragma unroll
            for (int rt = 0; rt < 4; ++rt) {     // A frags direct from global (L2-hot)
                const float* ap = Wck + (long)(rt * 16) * wCoStride
                                      + (long)(kk4 * 4) * wCiStride;
                a[rt].x = ap[0];
                a[rt].y = ap[wCiStride];
            }
            #pragma unroll
            for (int jc = 0; jc < 4; ++jc)       // B frags: one ds_load_b64 each
                bv[jc] = *(const v2f*)(Bb + (kk4 * 16 + jc) * 64);
            #pragma unroll
            for (int rt = 0; rt < 4; ++rt)
                #pragma unroll
                for (int jc = 0; jc < 4; ++jc)
                    acc[rt * 4 + jc] = __builtin_amdgcn_wmma_f32_16x16x4_f32(
                        false, a[rt], false, bv[jc], (short)0,
                        acc[rt * 4 + jc], false, false);
        }
    };

    // ---- software pipeline: load m+1 | wmma m | store m+1 | sync
    if (nchunk > 0) {
        load_chunk(0);
        store_chunk(0);
        __syncthreads();
        for (int m = 0; m < nchunk; ++m) {
            const int buf = m & 1;
            const bool more = (m + 1) < nchunk;
            if (more) load_chunk(m + 1);
            compute_chunk(m, buf);
            if (more) store_chunk(1 - buf);
            __syncthreads();
        }
    }

    // ---- epilogue: D layout -> lanes 0-15 rows M=r, lanes 16-31 rows M=r+8
    #pragma unroll
    for (int rt = 0; rt < 4; ++rt) {
        #pragma unroll
        for (int r = 0; r < 8; ++r) {
            const int co   = row0 + wr * 64 + rt * 16 + r + lhi * 8;
            const float s  = scale ? scale[co] : 1.0f;
            const float sh = shift ? shift[co] : 0.0f;
            float* op = OUTb + (long)co * outCoStride + wc * 64 + l16;
            #pragma unroll
            for (int jc = 0; jc < 4; ++jc)
                op[jc * 16] = acc[rt * 4 + jc][r] * s + sh;
        }
    }
}

// Analytic degrees of the (provably constant) temporal graph.
__device__ __forceinline__ float tg_deg(int t, int p) {
    if (p == 0) return (t == 0) ? 5.0f : ((t == VV - 1) ? 2.0f : 6.0f);
    if (p < 4)  return (t == 0) ? 1.0f : 2.0f;
    return 1.0f;
}

// out(t,p) = bg + xl(t,p)/deg + sum_{m in N} xl(m) * rsqrt(deg(m)*deg(t,p))
// N((t,0)) = {(t+1,j) j=0..3} u {(t-1,0)};  N((t,0<j<4)) = {(t-1,0)}
__global__ __launch_bounds__(256) void tg_gcn_fixup(
    const float* __restrict__ XL, const float* __restrict__ bg,
    float* __restrict__ O)
{
    const long idx = (long)blockIdx.x * 256 + threadIdx.x;   // over B*C*V*HW
    const int  p   = (int)(idx & (HW - 1));
    const long q   = idx >> 8;
    const int  t   = (int)(q & (VV - 1));
    const long bc  = q >> 7;                                 // b*C + co
    const int  co  = (int)(bc & (CC - 1));
    const long rowbase = bc * (long)(VV * HW);

    const float dn = tg_deg(t, p);
    float out = XL[idx] * (1.0f / dn) + bg[co];

    if (p == 0) {
        if (t + 1 < VV) {
            const float r0 = rsqrtf(dn * tg_deg(t + 1, 0));
            const float r1 = rsqrtf(dn * tg_deg(t + 1, 1));
            const float* nb = XL + rowbase + (long)(t + 1) * HW;
            out += nb[0] * r0;
            out += (nb[1] + nb[2] + nb[3]) * r1;
        }
        if (t >= 1)
            out += XL[rowbase + (long)(t - 1) * HW] * rsqrtf(dn * tg_deg(t - 1, 0));
    } else if (p < 4 && t >= 1) {
        out += XL[rowbase + (long)(t - 1) * HW] * rsqrtf(dn * tg_deg(t - 1, 0));
    }
    O[idx] = out;
}

extern "C" void kernel_launch(void* const* d_in, const int* in_sizes, int n_in,
                              void* d_out, int out_size, void* d_ws, size_t ws_size,
                              hipStream_t stream) {
    (void)in_sizes; (void)n_in; (void)out_size; (void)ws_size;
    const float* x  = (const float*)d_in[0];
    // d_in[1] = batch (==4, hardcoded)
    const float* Wd = (const float*)d_in[2];
    const float* s1 = (const float*)d_in[3];
    const float* t1 = (const float*)d_in[4];
    const float* Wg = (const float*)d_in[5];
    const float* bg = (const float*)d_in[6];
    const float* Wu = (const float*)d_in[7];
    const float* s2 = (const float*)d_in[8];
    const float* t2 = (const float*)d_in[9];
    float* out = (float*)d_out;

    float* xd = (float*)d_ws;            // (b,co,t,p)  134 MB
    float* XL = xd + (long)BB * BCVH;    // (b,co,t,p)  134 MB
    float* o4 = xd;                      // fixup output reuses xd buffer

    dim3 grid(BB * VV, 2), blk(256);

    // conv1: IN = x (tlen,c,hw); W = Wd (co,ci,k); OUT = xd (b,co,t,p)
    tg_gemm_tap<<<grid, blk, 0, stream>>>(x, Wd, s1, t1, xd,
        /*inT*/ CC * HW, /*inCi*/ HW, /*outT*/ HW, /*outCo*/ VV * HW,
        /*wCo*/ CC * 3, /*wCi*/ 3, /*wTap*/ 1, /*ntaps*/ 3, /*shift*/ -1);

    // GCN GEMM: XL(co,p) = Wg^T @ xd  (A[m][k] = Wg[ci][co] -> transposed strides)
    tg_gemm_tap<<<grid, blk, 0, stream>>>(xd, Wg, nullptr, nullptr, XL,
        /*inT*/ HW, /*inCi*/ VV * HW, /*outT*/ HW, /*outCo*/ VV * HW,
        /*wCo*/ 1, /*wCi*/ CC, /*wTap*/ 0, /*ntaps*/ 1, /*shift*/ 0);

    // GCN aggregation with analytic fixed-graph degrees
    tg_gcn_fixup<<<dim3((BB * BCVH) / 256), blk, 0, stream>>>(XL, bg, o4);

    // conv2: IN = o4 (b,ci,t,p); OUT = final (b*v+t, co, p)
    tg_gemm_tap<<<grid, blk, 0, stream>>>(o4, Wu, s2, t2, out,
        /*inT*/ HW, /*inCi*/ VV * HW, /*outT*/ CC * HW, /*outCo*/ HW,
        /*wCo*/ CC * 3, /*wCi*/ 3, /*wTap*/ 1, /*ntaps*/ 3, /*shift*/ -1);
}